// GINConv_59528246723311
// MI455X (gfx1250) — compile-verified
//
#include <hip/hip_runtime.h>

typedef __attribute__((ext_vector_type(2))) float v2f;
typedef __attribute__((ext_vector_type(8))) float v8f;

#define NN 50000
#define NE 800000
#define DF 64
#define DH 128
#define LDS_STRIDE 132   // 16 rows * 132 floats; bank-conflict-free for the access patterns below

// ---------------------------------------------------------------------------
// Kernel 1: acc[i] = (1 + eps) * feat[i]   (fuses the (1+eps)*feat term)
// ---------------------------------------------------------------------------
__global__ void gin_init(const float* __restrict__ feat,
                         const float* __restrict__ eps,
                         float* __restrict__ acc) {
  int i = blockIdx.x * blockDim.x + threadIdx.x;   // float4 granularity
  if (i < NN * DF / 4) {
    float s = 1.0f + eps[0];
    float4 v = ((const float4*)feat)[i];
    v.x *= s; v.y *= s; v.z *= s; v.w *= s;
    ((float4*)acc)[i] = v;
  }
}

// ---------------------------------------------------------------------------
// Kernel 2: scatter-add  acc[dst[e]][:] += feat[src[e]][:]
// 64 threads per edge, one f32 atomic per (edge,feature). Both acc and feat
// are L2-resident (12.8 MB each vs 192 MB L2), so atomics run at L2 rate.
// ---------------------------------------------------------------------------
__global__ void gin_scatter(const float* __restrict__ feat,
                            const int* __restrict__ src,
                            const int* __restrict__ dst,
                            float* __restrict__ acc) {
  long long t = (long long)blockIdx.x * blockDim.x + threadIdx.x;
  int e = (int)(t >> 6);
  int d = (int)(t & 63);
  if (e >= NE) return;
  int s = src[e];
  int q = dst[e];
  atomicAdd(acc + (long long)q * DF + d, feat[(long long)s * DF + d]);
}

// ---------------------------------------------------------------------------
// Kernel 3: out = relu(acc @ W1 + b1) @ W2 + b2  via V_WMMA_F32_16X16X4_F32.
// One wave (32 lanes) per 16-row tile; single-wave workgroup (barrier==s_nop).
//
// WMMA f32 16x16x4 lane layouts (ISA 7.12.2):
//   A (16x4):  lane L holds M = L&15;  VGPR0/1 = K = (L>=16 ? 2,3 : 0,1)
//   B (4x16):  lane L holds N = L&15;  VGPR0/1 = K = (L>=16 ? 2,3 : 0,1)
//   C/D:       VGPR r -> M = r + 8*(L>=16), N = L&15
// ---------------------------------------------------------------------------
__global__ __launch_bounds__(32) void gin_mlp(const float* __restrict__ acc,
                                              const float* __restrict__ W1,
                                              const float* __restrict__ b1,
                                              const float* __restrict__ W2,
                                              const float* __restrict__ b2,
                                              float* __restrict__ out) {
  __shared__ float lds[16 * LDS_STRIDE];

  const int tile = blockIdx.x;          // 0..3124 (50000/16), wave-uniform
  const int lane = threadIdx.x;         // 0..31
  const int half = lane >> 4;           // 0 or 1
  const int l16  = lane & 15;
  const int row0 = tile * 16;

  // ---- Preload A fragments for GEMM1 (K = 64 -> 16 k-steps of 4) ----
  v2f afrag[16];
  {
    const float* arow = acc + (long long)(row0 + l16) * DF;
#pragma unroll
    for (int kk = 0; kk < 16; ++kk) {
      int kb = kk * 4 + half * 2;
      afrag[kk].x = arow[kb];
      afrag[kk].y = arow[kb + 1];
    }
  }

  // ---- GEMM1: hidden(16x128) = relu(A @ W1 + b1), 8 N-tiles ----
#pragma unroll
  for (int nt = 0; nt < 8; ++nt) {
    const int n0 = nt * 16;
    v8f c = {};
#pragma unroll
    for (int kk = 0; kk < 16; ++kk) {
      int kr = kk * 4 + half * 2;
      v2f bf;
      bf.x = W1[kr * DH + n0 + l16];
      bf.y = W1[(kr + 1) * DH + n0 + l16];
      c = __builtin_amdgcn_wmma_f32_16x16x4_f32(false, afrag[kk], false, bf,
                                                (short)0, c, false, false);
    }
    float bias = b1[n0 + l16];
#pragma unroll
    for (int r = 0; r < 8; ++r) {
      float h = c[r] + bias;
      h = h > 0.0f ? h : 0.0f;
      lds[(r + half * 8) * LDS_STRIDE + n0 + l16] = h;
    }
  }

  __syncthreads();  // single-wave workgroup -> lowers to s_nop; orders LDS use

  // ---- GEMM2: out(16x64) = hidden @ W2 + b2; K = 128 -> 32 k-steps ----
#pragma unroll
  for (int nt = 0; nt < 4; ++nt) {
    const int n0 = nt * 16;
    v8f c = {};
#pragma unroll
    for (int kk = 0; kk < 32; ++kk) {
      int kr = kk * 4 + half * 2;
      v2f af, bf;
      af.x = lds[l16 * LDS_STRIDE + kr];
      af.y = lds[l16 * LDS_STRIDE + kr + 1];
      bf.x = W2[kr * DF + n0 + l16];
      bf.y = W2[(kr + 1) * DF + n0 + l16];
      c = __builtin_amdgcn_wmma_f32_16x16x4_f32(false, af, false, bf,
                                                (short)0, c, false, false);
    }
    float bias = b2[n0 + l16];
#pragma unroll
    for (int r = 0; r < 8; ++r) {
      out[(long long)(row0 + r + half * 8) * DF + n0 + l16] = c[r] + bias;
    }
  }
}

// ---------------------------------------------------------------------------
// Launch: init -> scatter -> MLP, all on `stream` (same-stream ordering).
// d_ws holds the 12.8 MB accumulator.
// ---------------------------------------------------------------------------
extern "C" void kernel_launch(void* const* d_in, const int* in_sizes, int n_in,
                              void* d_out, int out_size, void* d_ws, size_t ws_size,
                              hipStream_t stream) {
  const float* feat = (const float*)d_in[0];
  const int*   src  = (const int*)d_in[1];
  const int*   dst  = (const int*)d_in[2];
  const float* eps  = (const float*)d_in[3];
  const float* W1   = (const float*)d_in[4];
  const float* b1   = (const float*)d_in[5];
  const float* W2   = (const float*)d_in[6];
  const float* b2   = (const float*)d_in[7];
  float* acc = (float*)d_ws;            // 50000*64 floats = 12.8 MB scratch
  float* out = (float*)d_out;

  // init: 3.2M floats as 800K float4's
  {
    int n4 = NN * DF / 4;
    int blocks = (n4 + 255) / 256;
    gin_init<<<blocks, 256, 0, stream>>>(feat, eps, acc);
  }
  // scatter: 800K edges * 64 feats = 51.2M threads
  {
    long long total = (long long)NE * 64;
    int blocks = (int)((total + 255) / 256);
    gin_scatter<<<blocks, 256, 0, stream>>>(feat, src, dst, acc);
  }
  // MLP: one wave per 16-row tile
  {
    gin_mlp<<<NN / 16, 32, 0, stream>>>(acc, W1, b1, W2, b2, out);
  }
}